// LVSA_43920335569629
// MI455X (gfx1250) — compile-verified
//
#include <hip/hip_runtime.h>
#include <math.h>

#define NB   4
#define DIM  48
#define HW   512
#define PH   64
#define DSF  8

typedef __attribute__((ext_vector_type(2))) float v2f;
typedef __attribute__((ext_vector_type(8))) float v8f;

// ---------------------------------------------------------------------------
// Kernel 1: blockwise 8x8 max + unbiased variance over x (the first big pass).
// One workgroup per (b*c, block-row ph): 8 rows x 512 cols, 256 threads.
// Thread t: block-col pw = t/4, sub-lane sub = t%4 covers 2 cols x 8 rows.
// Fully coalesced float2 loads (256 threads cover one 2KB row exactly).
// ---------------------------------------------------------------------------
__global__ __launch_bounds__(256) void lvsa_pool_var(
    const float* __restrict__ x, float* __restrict__ pool, float* __restrict__ var)
{
    int g   = blockIdx.x;          // bc*PH + ph
    int ph  = g & (PH - 1);
    int bc  = g >> 6;              // b*DIM + c
    int tid = threadIdx.x;
    int pw  = tid >> 2;
    int sub = tid & 3;
    int col = pw * DSF + sub * 2;

    const float* base = x + ((size_t)bc * HW + (size_t)ph * DSF) * HW + col;

    float mx = -INFINITY, sum = 0.f, ss = 0.f;
#pragma unroll
    for (int r = 0; r < DSF; ++r) {
        float2 v = *(const float2*)(base + (size_t)r * HW);
        mx  = fmaxf(mx, fmaxf(v.x, v.y));
        sum += v.x + v.y;
        ss  += v.x * v.x + v.y * v.y;
    }
    // reduce across the 4 adjacent lanes that share a block (wave32 shuffles)
    mx  = fmaxf(mx, __shfl_xor(mx, 1, 32));
    sum += __shfl_xor(sum, 1, 32);
    ss  += __shfl_xor(ss, 1, 32);
    mx  = fmaxf(mx, __shfl_xor(mx, 2, 32));
    sum += __shfl_xor(sum, 2, 32);
    ss  += __shfl_xor(ss, 2, 32);

    if (sub == 0) {
        size_t o = ((size_t)bc * PH + ph) * PH + pw;
        pool[o] = mx;
        var[o]  = (ss - sum * sum * (1.0f / 64.0f)) * (1.0f / 63.0f);  // ddof=1
    }
}

// ---------------------------------------------------------------------------
// Kernel 2: gate map on the pooled grid.
//   T[i,p] = (dwconv3x3(pool)[i,p] + dw_b[i]) * alpha[i] + var[i,p] * belt[i]
//   G[o,p] = gelu( sum_i conv_w[o,i] * T[i,p] + conv_b[o] )   (exact GELU)
// One workgroup (128 thr = 4 waves) per (batch, pooled row y).
// The 48x48 channel mix runs on the matrix pipe: V_WMMA_F32_16X16X4_F32,
// K=48 in 12 steps, 3 output chunks of 16 rows (fp32-exact).
// ---------------------------------------------------------------------------
__global__ __launch_bounds__(128) void lvsa_gate(
    const float* __restrict__ pool, const float* __restrict__ var,
    const float* __restrict__ conv_w, const float* __restrict__ conv_b,
    const float* __restrict__ dw_w,   const float* __restrict__ dw_b,
    const float* __restrict__ alpha,  const float* __restrict__ belt,
    float* __restrict__ gate)
{
    __shared__ float Wl[DIM * DIM];    // conv_w  [o][i]
    __shared__ float DWl[DIM * 9];     // dw_w    [i][3][3]
    __shared__ float Al[DIM], Bel[DIM], DBl[DIM], CBl[DIM];
    __shared__ float Tl[DIM][PH];      // gate input matrix T[i][x]

    int t = threadIdx.x;
    int b = blockIdx.x >> 6;
    int y = blockIdx.x & 63;

    for (int k = t; k < DIM * DIM; k += 128) Wl[k]  = conv_w[k];
    for (int k = t; k < DIM * 9;   k += 128) DWl[k] = dw_w[k];
    if (t < DIM) { Al[t] = alpha[t]; Bel[t] = belt[t]; DBl[t] = dw_b[t]; CBl[t] = conv_b[t]; }
    __syncthreads();

    const float* pb = pool + (size_t)b * DIM * PH * PH;
    const float* vb = var  + (size_t)b * DIM * PH * PH;

    // cooperative depthwise conv + affine mix into LDS (48*64 values / 128 thr)
    for (int idx = t; idx < DIM * PH; idx += 128) {
        int i  = idx >> 6;
        int xx = idx & 63;
        const float* pc = pb + (size_t)i * PH * PH;
        float acc = 0.f;
#pragma unroll
        for (int dy = -1; dy <= 1; ++dy) {
            int yy = y + dy;
            if ((unsigned)yy < (unsigned)PH) {
                const float* row = pc + yy * PH;
#pragma unroll
                for (int dx = -1; dx <= 1; ++dx) {
                    int xc = xx + dx;
                    if ((unsigned)xc < (unsigned)PH)
                        acc += DWl[i * 9 + (dy + 1) * 3 + (dx + 1)] * row[xc];
                }
            }
        }
        float xs = acc + DBl[i];
        Tl[i][xx] = xs * Al[i] + vb[(size_t)i * PH * PH + (size_t)y * PH + xx] * Bel[i];
    }
    __syncthreads();

    // WMMA channel mix: wave w owns pixels [w*16, w*16+16)
    int wave = t >> 5;
    int lane = t & 31;
    int hlf  = lane >> 4;      // which 16-lane half
    int m    = lane & 15;
    int pbase = wave * 16;

    v8f acc0 = {0,0,0,0,0,0,0,0};
    v8f acc1 = acc0, acc2 = acc0;

#pragma unroll
    for (int k = 0; k < DIM; k += 4) {
        // B 4x16 f32: VGPR0 = K {0 | 2} rows, VGPR1 = K {1 | 3} rows, N = lane%16
        v2f bf;
        bf.x = Tl[k     + 2 * hlf][pbase + m];
        bf.y = Tl[k + 1 + 2 * hlf][pbase + m];
        // A 16x4 f32: lane m holds row M=m; VGPR0 = K {0|2}, VGPR1 = K {1|3}
        v2f a0, a1, a2;
        a0.x = Wl[( 0 + m) * DIM + k     + 2 * hlf];
        a0.y = Wl[( 0 + m) * DIM + k + 1 + 2 * hlf];
        a1.x = Wl[(16 + m) * DIM + k     + 2 * hlf];
        a1.y = Wl[(16 + m) * DIM + k + 1 + 2 * hlf];
        a2.x = Wl[(32 + m) * DIM + k     + 2 * hlf];
        a2.y = Wl[(32 + m) * DIM + k + 1 + 2 * hlf];
        acc0 = __builtin_amdgcn_wmma_f32_16x16x4_f32(false, a0, false, bf, (short)0, acc0, false, false);
        acc1 = __builtin_amdgcn_wmma_f32_16x16x4_f32(false, a1, false, bf, (short)0, acc1, false, false);
        acc2 = __builtin_amdgcn_wmma_f32_16x16x4_f32(false, a2, false, bf, (short)0, acc2, false, false);
    }

    // D layout: VGPR v, lane -> (o_rel = v + 8*hlf, pixel = pbase + m)
    float* gb = gate + (size_t)b * DIM * PH * PH + (size_t)y * PH + (pbase + m);
#pragma unroll
    for (int v = 0; v < 8; ++v) {
        int orel = v + 8 * hlf;
        float g0 = acc0[v] + CBl[orel];
        float g1 = acc1[v] + CBl[orel + 16];
        float g2 = acc2[v] + CBl[orel + 32];
        g0 = 0.5f * g0 * (1.0f + erff(g0 * 0.70710678118654752f));
        g1 = 0.5f * g1 * (1.0f + erff(g1 * 0.70710678118654752f));
        g2 = 0.5f * g2 * (1.0f + erff(g2 * 0.70710678118654752f));
        gb[(size_t)(orel     ) * PH * PH] = g0;
        gb[(size_t)(orel + 16) * PH * PH] = g1;
        gb[(size_t)(orel + 32) * PH * PH] = g2;
    }
}

// ---------------------------------------------------------------------------
// Kernel 3: out = x * upsample8(gate). Pure float4 streaming pass; the 3MB
// gate map stays hot in L1/L2 (each gate value reused 16x along a row pair).
// ---------------------------------------------------------------------------
__global__ __launch_bounds__(256) void lvsa_apply(
    const float* __restrict__ x, const float* __restrict__ gate, float* __restrict__ out)
{
    size_t f = (size_t)blockIdx.x * blockDim.x + threadIdx.x;   // float4 index
    int    w4   = (int)(f & 127);          // 128 float4 per 512-wide row
    size_t rest = f >> 7;
    int    h    = (int)(rest & 511);
    size_t cb   = rest >> 9;               // b*DIM + c

    float g = gate[(cb * PH + (size_t)(h >> 3)) * PH + (w4 >> 1)];

    float4 xv = ((const float4*)x)[f];
    float4 o;
    o.x = xv.x * g; o.y = xv.y * g; o.z = xv.z * g; o.w = xv.w * g;
    ((float4*)out)[f] = o;
}

// ---------------------------------------------------------------------------
extern "C" void kernel_launch(void* const* d_in, const int* in_sizes, int n_in,
                              void* d_out, int out_size, void* d_ws, size_t ws_size,
                              hipStream_t stream)
{
    const float* x      = (const float*)d_in[0];
    const float* conv_w = (const float*)d_in[1];
    const float* conv_b = (const float*)d_in[2];
    const float* dw_w   = (const float*)d_in[3];
    const float* dw_b   = (const float*)d_in[4];
    const float* alpha  = (const float*)d_in[5];
    const float* belt   = (const float*)d_in[6];
    float* out = (float*)d_out;

    const size_t mapElems = (size_t)NB * DIM * PH * PH;    // 786432
    float* pool = (float*)d_ws;
    float* var  = pool + mapElems;
    float* gate = var  + mapElems;

    // Pass 1: pooled max + unbiased variance (reads x once)
    lvsa_pool_var<<<NB * DIM * PH, 256, 0, stream>>>(x, pool, var);

    // Pass 2: gate map via dwconv + WMMA channel mix + GELU
    lvsa_gate<<<NB * PH, 128, 0, stream>>>(pool, var, conv_w, conv_b,
                                           dw_w, dw_b, alpha, belt, gate);

    // Pass 3: out = x * upsample(gate)  (reads x again, writes out)
    const size_t n4 = (size_t)NB * DIM * HW * HW / 4;      // 12,582,912
    lvsa_apply<<<(unsigned)(n4 / 256), 256, 0, stream>>>(x, gate, out);
}